// Poolpointsinterp_8538394984585
// MI455X (gfx1250) — compile-verified
//
#include <hip/hip_runtime.h>

typedef __attribute__((ext_vector_type(2))) float v2f;
typedef __attribute__((ext_vector_type(8))) float v8f;

#define N_  8
#define C_  256
#define H_  128
#define W_  128
#define P_  100000
#define SCALE_ 0.25f
#define WPB 8           // waves per block
#define CPW 4           // 16-channel chunks per wave (64 channels)

// One wave handles one point x 64 channels (4 WMMA chunks of 16).
// WMMA f32 16x16x4: A = point's 4 bilinear weights replicated over rows,
// B = 4x16 gathered taps (lane half = y row, vgpr0/1 = x_low/x_high tap).
// D rows are all identical, so lane half 0 serves even chunks and lane half 1
// serves odd chunks -> full-EXEC stores, no predication.
__global__ __launch_bounds__(256) void Poolpointsinterp_kernel(
    const float* __restrict__ feat,   // (N, C, H, W) f32
    const float* __restrict__ rois,   // (P, 3) f32: [batch, x, y]
    float* __restrict__ out)          // (P, C) f32
{
  const int  lane    = threadIdx.x & 31;
  const int  wave    = threadIdx.x >> 5;
  const int  p       = blockIdx.x * WPB + wave;
  const int  col     = lane & 15;
  const bool lowHalf = lane < 16;
  // channel this lane's column maps to in chunk 0 of this wave's 64-ch slab
  const int  c0      = blockIdx.y * (CPW * 16) + col;

  // ---- per-point setup (done once per 64 channels) ----
  const float rb = rois[p * 3 + 0];
  float x = rois[p * 3 + 1] * SCALE_;
  float y = rois[p * 3 + 2] * SCALE_;
  const int b = (int)rb;

  const bool valid = (y >= -1.0f) && (y <= (float)H_) &&
                     (x >= -1.0f) && (x <= (float)W_);

  y = fmaxf(y, 0.0f);
  x = fmaxf(x, 0.0f);
  int y_low = (int)floorf(y);
  int x_low = (int)floorf(x);
  const bool y_edge = y_low >= H_ - 1;
  const bool x_edge = x_low >= W_ - 1;
  y_low = y_edge ? (H_ - 1) : y_low;
  x_low = x_edge ? (W_ - 1) : x_low;
  const int y_high = y_edge ? y_low : y_low + 1;
  const int x_high = x_edge ? x_low : x_low + 1;
  y = y_edge ? (float)y_low : y;
  x = x_edge ? (float)x_low : x;
  const float ly = y - (float)y_low, lx = x - (float)x_low;
  const float hy = 1.0f - ly,        hx = 1.0f - lx;

  // A(16x4): lanes 0-15 hold K=0,1 -> (w1,w2); lanes 16-31 hold K=2,3 ->
  // (w3,w4). Scale by valid? so invalid points produce exact 0 (addresses
  // below are always clamped in-bounds, features are finite).
  float wy = lowHalf ? hy : ly;
  wy = valid ? wy : 0.0f;
  v2f a;
  a.x = wy * hx;
  a.y = wy * lx;

  // ---- gathered B operands: 2 loads/lane/chunk, chunk stride = 1 MB imm ----
  const int y_sel = lowHalf ? y_low : y_high;
  const float* __restrict__ rowp =
      feat + (((size_t)b * C_ + c0) * H_ + y_sel) * W_;
  const float* __restrict__ pLo = rowp + x_low;
  const float* __restrict__ pHi = rowp + x_high;

  const int CH = 16 * H_ * W_;  // 262144 floats: 16-channel stride (1 MB)
  v2f b0; b0.x = pLo[0 * CH]; b0.y = pHi[0 * CH];
  v2f b1; b1.x = pLo[1 * CH]; b1.y = pHi[1 * CH];
  v2f b2; b2.x = pLo[2 * CH]; b2.y = pHi[2 * CH];
  v2f b3; b3.x = pLo[3 * CH]; b3.y = pHi[3 * CH];

  const v8f z = {};
  v8f d0 = __builtin_amdgcn_wmma_f32_16x16x4_f32(false, a, false, b0, (short)0, z, false, false);
  v8f d1 = __builtin_amdgcn_wmma_f32_16x16x4_f32(false, a, false, b1, (short)0, z, false, false);
  v8f d2 = __builtin_amdgcn_wmma_f32_16x16x4_f32(false, a, false, b2, (short)0, z, false, false);
  v8f d3 = __builtin_amdgcn_wmma_f32_16x16x4_f32(false, a, false, b3, (short)0, z, false, false);

  // ---- full-EXEC NT stores: low half stores even chunk, high half odd ----
  // lane L<16 : d_{2j}[0] = row M=0, N=L      -> channel c0 + 32j
  // lane L>=16: d_{2j+1}[0] = row M=8, N=L-16 -> channel c0 + 16 + 32j
  const float v01 = lowHalf ? d0[0] : d1[0];
  const float v23 = lowHalf ? d2[0] : d3[0];
  float* o = out + (size_t)p * C_ + c0 + (lowHalf ? 0 : 16);
  __builtin_nontemporal_store(v01, o);
  __builtin_nontemporal_store(v23, o + 32);
}

extern "C" void kernel_launch(void* const* d_in, const int* in_sizes, int n_in,
                              void* d_out, int out_size, void* d_ws, size_t ws_size,
                              hipStream_t stream) {
  const float* feat = (const float*)d_in[0];   // 8*256*128*128 f32
  const float* rois = (const float*)d_in[1];   // 100000*3 f32
  float* out = (float*)d_out;                  // 100000*256 f32

  // x-major dispatch: all points for one 64-channel slab run together;
  // slab working set = 8*64*128*128*4 = 32 MB < 192 MB L2.
  dim3 grid(P_ / WPB, C_ / (16 * CPW), 1);     // 12500 x 4
  Poolpointsinterp_kernel<<<grid, 256, 0, stream>>>(feat, rois, out);
}